// BAPClassifier_69552700391677
// MI455X (gfx1250) — compile-verified
//
#include <hip/hip_runtime.h>
#include <cmath>

// ---------------------------------------------------------------------------
// BAP classifier for MI455X (gfx1250, wave32).
// Bandwidth-bound problem (~210 MB HBM, ~6.6 GFLOP) -> fp32 WMMA 16x16x4.
// x (103MB) and Wc (104MB) stream from HBM once; feats/Wa/A/partials are
// L2-resident (192MB). Roofline ~9us @ 23.3 TB/s; WMMA f32 is ample.
// ---------------------------------------------------------------------------

typedef __attribute__((ext_vector_type(2))) float v2f;
typedef __attribute__((ext_vector_type(8))) float v8f;

#define WMMA_F32(a, b, c) \
  __builtin_amdgcn_wmma_f32_16x16x4_f32(false, (a), false, (b), (short)0, (c), false, false)

#define B_   64
#define C_   2048
#define HW_  196
#define M_   32
#define NC_  396
#define K3_  (M_ * C_)   // 65536
#define NPAD_ 400        // 25 n-tiles of 16
#define NTILES_ 25
#define KCHUNK_ 1024
#define NKCH_   64       // 65536 / 1024

__device__ __forceinline__ float fast_sigmoid(float v) {
  // v_exp_f32 + v_add + v_rcp_f32 (no IEEE divide chain)
  return __builtin_amdgcn_rcpf(1.0f + __expf(-v));
}

// ---------------------------------------------------------------------------
// Kernel 1: A[b,m,hw] = sigmoid( sum_c x[b,c,hw] * Wa[m,c] + ba[m] )
// 4 waves/block; each wave owns one (b, hw-tile).
// Matrix-M = hw (16), Matrix-N = m (2 tiles), K = C.
// ---------------------------------------------------------------------------
__global__ __launch_bounds__(128) void k_attn(const float* __restrict__ x,
                                              const float* __restrict__ Wa,
                                              const float* __restrict__ ba,
                                              float* __restrict__ A_ws) {
  const int wave = threadIdx.x >> 5;
  const int hwt  = blockIdx.x * 4 + wave;  // 0..15, 13 used
  if (hwt >= 13) return;                   // wave-uniform exit
  const int b    = blockIdx.y;             // 0..63
  const int lane = threadIdx.x & 31;
  const int half = lane >> 4;
  const int r    = lane & 15;
  const int hw0  = hwt * 16;
  const int hwr  = min(hw0 + r, HW_ - 1);  // clamp loads; bogus rows discarded at store

  v8f acc0 = {}; v8f acc1 = {};
  const float* xb = x + (size_t)b * C_ * HW_;

  for (int k0 = 0; k0 < C_; k0 += 4) {
    const int k = k0 + 2 * half;
    // A fragment: A[row=hw][k] = x[b][k][hw]  (strided over c, coalesced over hw)
    v2f a;
    a.x = xb[(size_t)k * HW_ + hwr];
    a.y = xb[(size_t)(k + 1) * HW_ + hwr];
    // B fragments: B[k][col=m] = Wa[m][k]; Wa is tiny -> L2-resident
    v2f b0 = *(const v2f*)(Wa + (size_t)(r)      * C_ + k);
    v2f b1 = *(const v2f*)(Wa + (size_t)(16 + r) * C_ + k);
    acc0 = WMMA_F32(a, b0, acc0);
    acc1 = WMMA_F32(a, b1, acc1);
  }

#pragma unroll
  for (int i = 0; i < 8; ++i) {
    const int hw = hw0 + i + (half ? 8 : 0);     // D row (hw dim)
    if (hw < HW_) {
      A_ws[((size_t)b * M_ + r)      * HW_ + hw] = fast_sigmoid(acc0[i] + ba[r]);
      A_ws[((size_t)b * M_ + 16 + r) * HW_ + hw] = fast_sigmoid(acc1[i] + ba[16 + r]);
    }
  }
}

// ---------------------------------------------------------------------------
// Kernel 2: feats[b, m*C + c] = (1/196) * sum_hw x[b,c,hw] * A[b,m,hw]
// 4 waves/block; each wave owns one (b, c-tile).
// Matrix-M = m (2 tiles), Matrix-N = c (16), K = HW.
// ---------------------------------------------------------------------------
__global__ __launch_bounds__(128) void k_bap(const float* __restrict__ x,
                                             const float* __restrict__ A_ws,
                                             float* __restrict__ feats) {
  const int wave = threadIdx.x >> 5;
  const int ct   = blockIdx.x * 4 + wave;  // 0..127 exact (grid.x = 32)
  const int b    = blockIdx.y;             // 0..63
  const int lane = threadIdx.x & 31;
  const int half = lane >> 4;
  const int r    = lane & 15;
  const int c0   = ct * 16;

  v8f acc0 = {}; v8f acc1 = {};
  const float* Ab = A_ws + (size_t)b * M_ * HW_;
  const float* xb = x + ((size_t)b * C_ + c0) * HW_;

  for (int k0 = 0; k0 < HW_; k0 += 4) {  // 196 = 49*4, exact
    const int k = k0 + 2 * half;
    v2f a0 = *(const v2f*)(Ab + (size_t)(r)      * HW_ + k);   // A rows = m (tile 0)
    v2f a1 = *(const v2f*)(Ab + (size_t)(16 + r) * HW_ + k);   // m (tile 1)
    v2f bb = *(const v2f*)(xb + (size_t)r * HW_ + k);          // B[k][col=c] = x[b][c][k]
    acc0 = WMMA_F32(a0, bb, acc0);
    acc1 = WMMA_F32(a1, bb, acc1);
  }

  const float s = 1.0f / (float)HW_;
#pragma unroll
  for (int i = 0; i < 8; ++i) {
    const int mr = i + (half ? 8 : 0);   // D row (m dim within tile)
    feats[((size_t)b * M_ + mr)      * C_ + c0 + r] = acc0[i] * s;
    feats[((size_t)b * M_ + 16 + mr) * C_ + c0 + r] = acc1[i] * s;
  }
}

// ---------------------------------------------------------------------------
// Kernel 3: split-K classifier GEMM partials.
// part[kc][b][n] = sum_{k in chunk kc} feats[b][k] * Wc[n][k]
// 8 waves/block, each wave owns ONE n-tile (Wc streamed from HBM exactly once)
// and all four 16-row b-tiles (feats re-reads hit 192MB L2).
// ---------------------------------------------------------------------------
__global__ __launch_bounds__(256) void k_fc_partial(const float* __restrict__ feats,
                                                    const float* __restrict__ Wc,
                                                    float* __restrict__ part) {
  const int wave = threadIdx.x >> 5;
  const int nt   = blockIdx.x * 8 + wave;   // n-tile id
  if (nt >= NTILES_) return;                // wave-uniform exit
  const int kc   = blockIdx.y;              // 0..63
  const int lane = threadIdx.x & 31;
  const int half = lane >> 4;
  const int r    = lane & 15;
  const int n0   = nt * 16;

  const bool nvalid = (n0 + r) < NC_;
  const int  nrow   = min(n0 + r, NC_ - 1);          // clamped, select applied below
  const float* wrow = Wc + (size_t)nrow * K3_;
  const int kbase   = kc * KCHUNK_;

  v8f acc[4] = {{}, {}, {}, {}};

  for (int k0 = 0; k0 < KCHUNK_; k0 += 4) {
    const int k = kbase + k0 + 2 * half;
    v2f bb = *(const v2f*)(wrow + k);                // B[k][col=n] = Wc[n][k]
    bb.x = nvalid ? bb.x : 0.0f;                     // v_cndmask: EXEC stays all-1
    bb.y = nvalid ? bb.y : 0.0f;
#pragma unroll
    for (int mt = 0; mt < 4; ++mt) {
      v2f a = *(const v2f*)(feats + (size_t)(mt * 16 + r) * K3_ + k);  // A rows = b
      acc[mt] = WMMA_F32(a, bb, acc[mt]);
    }
  }

#pragma unroll
  for (int mt = 0; mt < 4; ++mt) {
#pragma unroll
    for (int i = 0; i < 8; ++i) {
      const int brow = mt * 16 + i + (half ? 8 : 0);
      part[((size_t)kc * B_ + brow) * NPAD_ + n0 + r] = acc[mt][i];
    }
  }
}

// ---------------------------------------------------------------------------
// Kernel 4: deterministic split-K reduction + bias.
// ---------------------------------------------------------------------------
__global__ __launch_bounds__(256) void k_reduce(const float* __restrict__ part,
                                               const float* __restrict__ bc,
                                               float* __restrict__ out) {
  const int idx = blockIdx.x * blockDim.x + threadIdx.x;
  if (idx >= B_ * NC_) return;
  const int b = idx / NC_;
  const int n = idx - b * NC_;
  float s = bc[n];
#pragma unroll 4
  for (int kc = 0; kc < NKCH_; ++kc)
    s += part[((size_t)kc * B_ + b) * NPAD_ + n];
  out[idx] = s;
}

// ---------------------------------------------------------------------------
extern "C" void kernel_launch(void* const* d_in, const int* in_sizes, int n_in,
                              void* d_out, int out_size, void* d_ws, size_t ws_size,
                              hipStream_t stream) {
  const float* x  = (const float*)d_in[0];   // [64, 2048, 14, 14]
  const float* Wa = (const float*)d_in[1];   // [32, 2048]
  const float* ba = (const float*)d_in[2];   // [32]
  const float* Wc = (const float*)d_in[3];   // [396, 65536]
  const float* bc = (const float*)d_in[4];   // [396]
  float* out = (float*)d_out;                // [64, 396]

  // workspace layout (floats): A (1.6MB) | feats (16.8MB) | partials (6.6MB)
  float* A_ws    = (float*)d_ws;                     // 64*32*196   = 401408
  float* feats   = A_ws + (size_t)B_ * M_ * HW_;     // 64*32*2048  = 4194304
  float* part_ws = feats + (size_t)B_ * K3_;         // 64*64*400   = 1638400

  k_attn      <<<dim3(4, B_),   128, 0, stream>>>(x, Wa, ba, A_ws);
  k_bap       <<<dim3(32, B_),  128, 0, stream>>>(x, A_ws, feats);
  k_fc_partial<<<dim3(4, NKCH_), 256, 0, stream>>>(feats, Wc, part_ws);
  k_reduce    <<<(B_ * NC_ + 255) / 256, 256, 0, stream>>>(part_ws, bc, out);
}